// RasterScanner_25898652795270
// MI455X (gfx1250) — compile-verified
//
#include <hip/hip_runtime.h>

// ---------------- problem constants (match reference) ----------------
constexpr int kB = 4;
constexpr int kH = 640;
constexpr int kW = 640;
constexpr int kHP = 641;                 // padded height
constexpr int kWP = 641;                 // padded width
constexpr int kPlane = 641 * 641;        // padded plane elements
constexpr int kNPix = kH * kW;           // 409600
constexpr int kThreads = 256;
constexpr int kTPT = 16;                 // pixels per thread
constexpr int kPixPerBlk = kThreads * kTPT;        // 4096
constexpr int kBlksPerPlane = kNPix / kPixPerBlk;  // 100
constexpr int kNBlocks = kB * kBlksPerPlane;       // 400
constexpr int kPartStride = 128;
constexpr int kIters = 64;
constexpr int kWinRows = 9;              // padded rows a block can touch
constexpr int kWin = kWinRows * kWP;     // 5769 window elements

typedef __attribute__((ext_vector_type(2))) float v2f;
typedef __attribute__((ext_vector_type(8))) float v8f;
typedef unsigned int u32;
typedef __attribute__((ext_vector_type(4))) u32 v4u;
typedef __attribute__((ext_vector_type(8))) u32 v8u;

__device__ __forceinline__ float resprime(const float* __restrict__ res,
                                          const float* __restrict__ M,
                                          const float* __restrict__ E, int o) {
  return fmaf(res[o], M[o], E[o]);  // res' = res*M + E
}

// ---- TDM: async 2D tile load (rows x 641 f32) from global into LDS ----
// D# group0 (4 SGPRs) + group1 (8 SGPRs), VADDR2/3 = NULL (2D tensor).
__device__ __forceinline__ void tdm_load2d(const float* gsrc, u32 ldsOff,
                                           int rowsAvail) {
  unsigned long long ga = (unsigned long long)(size_t)gsrc;
  v4u g0;
  g0[0] = 1u;                                   // count=1, no gather
  g0[1] = ldsOff;                               // lds_addr (bytes)
  g0[2] = (u32)ga;                              // global_addr[31:0]
  g0[3] = ((u32)(ga >> 32) & 0x01FFFFFFu) | (2u << 30);  // addr[56:32], type=2
  v8u g1;
  g1[0] = 2u << 16;                             // data_size=4B, no flags
  g1[1] = ((u32)kWP & 0xFFFFu) << 16;           // tensor_dim0 low16 (=641)
  g1[2] = ((u32)rowsAvail & 0xFFFFu) << 16;     // dim0 hi16=0 | tensor_dim1 low16
  g1[3] = ((u32)kWP) << 16;                     // dim1 hi16=0 | tile_dim0=641
  g1[4] = (u32)kWinRows;                        // tile_dim1=9, tile_dim2=0
  g1[5] = (u32)kWP;                             // tensor_dim0_stride low32 (=641)
  g1[6] = 0u;                                   // stride0 hi16 | stride1 low16
  g1[7] = 0u;                                   // stride1 hi32
  asm volatile("tensor_load_to_lds %0, %1" : : "s"(g0), "s"(g1) : "memory");
}

// ---------------- init: build padded res0/M0/E0, clear flags ----------------
__global__ void k_init(const float* __restrict__ img, float* __restrict__ res,
                       float* __restrict__ M, float* __restrict__ E,
                       int* __restrict__ flags) {
  long total = (long)kB * kPlane;
  for (long idx = (long)blockIdx.x * blockDim.x + threadIdx.x; idx < total;
       idx += (long)gridDim.x * blockDim.x) {
    int b = (int)(idx / kPlane);
    int r = (int)(idx - (long)b * kPlane);
    int i = r / kWP;
    int j = r - i * kWP;
    float v = 0.f;
    if (i > 0 && j > 0) v = img[(long)b * kNPix + (i - 1) * kW + (j - 1)];
    res[idx] = v;
    M[idx] = 1.f;
    E[idx] = 0.f;
  }
  if (blockIdx.x == 0 && threadIdx.x == 0) { flags[0] = 0; flags[1] = 0; }
}

// ---------------- phase 1: per-block sums of pos ----------------
__global__ void k_partials(const float* __restrict__ res, const float* __restrict__ M,
                           const float* __restrict__ E, const float* __restrict__ img,
                           float* __restrict__ partials, const int* __restrict__ flags) {
  if (flags[0]) return;  // converged
  __shared__ float sred[kThreads];
  int blk = blockIdx.x % kBlksPerPlane;
  int p = blockIdx.x / kBlksPerPlane;
  int base = p * kPlane;
  const float* dp = img + (long)p * kNPix;
  int pixBase = blk * kPixPerBlk + threadIdx.x * kTPT;
  __builtin_prefetch(dp + pixBase, 0, 1);  // global_prefetch_b8
  float s = 0.f;
#pragma unroll
  for (int q = 0; q < kTPT; ++q) {
    int e = pixBase + q;
    int y = e / kW;
    int x = e - y * kW;
    int o = base + y * kWP + x;
    float a = resprime(res, M, E, o);
    float t = resprime(res, M, E, o + 1);
    float l = resprime(res, M, E, o + kWP);
    float all = a + t + l;
    s += (all == 0.f) ? dp[e] : 0.f;
  }
  sred[threadIdx.x] = s;
  __syncthreads();
  for (int off = kThreads / 2; off > 0; off >>= 1) {
    if ((int)threadIdx.x < off) sred[threadIdx.x] += sred[threadIdx.x + off];
    __syncthreads();
  }
  if (threadIdx.x == 0) partials[p * kPartStride + blk] = sred[0];
}

// ---------------- phase 2: exclusive scan of 100 block sums per plane ----------------
__global__ void k_scanblk(const float* __restrict__ partials,
                          float* __restrict__ blkoffs, int* __restrict__ flags) {
  if (flags[0]) return;
  if (threadIdx.x == 0) flags[1] = 0;  // reset "changed"
  int lane = threadIdx.x & 31;
  int p = threadIdx.x >> 5;  // 4 waves, one plane per wave
  float ex[4];
  float s = 0.f;
#pragma unroll
  for (int q = 0; q < 4; ++q) {
    int idx = lane * 4 + q;
    float v = (idx < kBlksPerPlane) ? partials[p * kPartStride + idx] : 0.f;
    ex[q] = s;
    s += v;
  }
  float inc = s;
#pragma unroll
  for (int d = 1; d < 32; d <<= 1) {
    float u = __shfl_up(inc, d, 32);
    if (lane >= d) inc += u;
  }
  float lexcl = inc - s;
#pragma unroll
  for (int q = 0; q < 4; ++q)
    blkoffs[p * kPartStride + lane * 4 + q] = lexcl + ex[q];
}

// ---------------- phase 3: TDM-staged fused scan + stencil update ----------------
__global__ void k_update(const float* __restrict__ res, const float* __restrict__ M,
                         const float* __restrict__ E, float* __restrict__ resN,
                         float* __restrict__ MN, float* __restrict__ EN,
                         const float* __restrict__ img,
                         const float* __restrict__ blkoffs, int* __restrict__ flags) {
  if (flags[0]) {
    // Converged: carry frozen state through the ping-pong.
    long n = (long)kB * kPlane;
    for (long i = (long)blockIdx.x * blockDim.x + threadIdx.x; i < n;
         i += (long)gridDim.x * blockDim.x) {
      resN[i] = res[i];
      MN[i] = M[i];
      EN[i] = E[i];
    }
    return;
  }
  __shared__ float s_rp[kWin];   // res window, turned into res' in place
  __shared__ float s_m[kWin];    // M window
  __shared__ float tot_lds[256];
  __shared__ float tmat[256];
  __shared__ float rowoff[16];
  __shared__ int anydiff;
  if (threadIdx.x == 0) anydiff = 0;

  int blk = blockIdx.x % kBlksPerPlane;
  int p = blockIdx.x / kBlksPerPlane;
  int base = p * kPlane;
  const float* dp = img + (long)p * kNPix;
  int blkPixBase = blk * kPixPerBlk;
  int y0 = blkPixBase / kW;                       // first padded row touched
  int rowsAvail = kHP - y0;
  if (rowsAvail > kWinRows) rowsAvail = kWinRows; // TDM zero-fills past dim1

  // --- async TDM stage of res & M windows (wave 0 issues; EXEC ignored) ---
  if (threadIdx.x == 0) {
    tdm_load2d(res + base + y0 * kWP, (u32)(size_t)(void*)s_rp, rowsAvail);
    tdm_load2d(M + base + y0 * kWP, (u32)(size_t)(void*)s_m, rowsAvail);
    __builtin_amdgcn_s_wait_tensorcnt(0);
  }
  __syncthreads();

  // --- res' = res*M + E computed once per window element, in LDS ---
  {
    const float* Ew = E + base + y0 * kWP;
    int n = rowsAvail * kWP;
    for (int i = threadIdx.x; i < n; i += kThreads)
      s_rp[i] = fmaf(s_rp[i], s_m[i], Ew[i]);
  }
  __syncthreads();

  int pixBase = blkPixBase + threadIdx.x * kTPT;
  __builtin_prefetch(dp + pixBase, 0, 1);

  // pass 1: seed mask + thread-local total (neighbors from LDS)
  float pos[kTPT];
  float s = 0.f;
#pragma unroll
  for (int q = 0; q < kTPT; ++q) {
    int e = pixBase + q;
    int y = e / kW;
    int x = e - y * kW;
    int lo = (y - y0) * kWP + x;
    float a = s_rp[lo];
    float t = s_rp[lo + 1];
    float l = s_rp[lo + kWP];
    float all = a + t + l;
    float pv = (all == 0.f) ? dp[e] : 0.f;
    pos[q] = pv;
    s += pv;
  }
  tot_lds[threadIdx.x] = s;
  __syncthreads();

  // WMMA matrix scan: T = X * U (upper-triangular ones), 4x K=4 chained
  if (threadIdx.x < 32) {
    int m = threadIdx.x & 15;
    int hi = threadIdx.x >> 4;
    v8f c = {0.f, 0.f, 0.f, 0.f, 0.f, 0.f, 0.f, 0.f};
#pragma unroll
    for (int kb = 0; kb < 4; ++kb) {
      int col0 = 4 * kb + 2 * hi;
      v2f A;
      A.x = tot_lds[m * 16 + col0];
      A.y = tot_lds[m * 16 + col0 + 1];
      int krow = 4 * kb + hi;
      v2f Bv;
      Bv.x = (krow <= m) ? 1.f : 0.f;
      Bv.y = (krow + 2 <= m) ? 1.f : 0.f;
      c = __builtin_amdgcn_wmma_f32_16x16x4_f32(false, A, false, Bv, (short)0, c,
                                                false, false);
    }
    int mb = hi * 8;
#pragma unroll
    for (int r = 0; r < 8; ++r) tmat[(mb + r) * 16 + m] = c[r];
  }
  __syncthreads();
  if (threadIdx.x < 16) {
    float rs = tmat[threadIdx.x * 16 + 15];
    float inc = rs;
#pragma unroll
    for (int d = 1; d < 16; d <<= 1) {
      float u = __shfl_up(inc, d, 32);
      if ((int)threadIdx.x >= d) inc += u;
    }
    rowoff[threadIdx.x] = inc - rs;
  }
  __syncthreads();

  float run = tmat[threadIdx.x] + rowoff[threadIdx.x >> 4] - s +
              blkoffs[p * kPartStride + blk];

  bool diff = false;
#pragma unroll
  for (int q = 0; q < kTPT; ++q) {
    int e = pixBase + q;
    int y = e / kW;
    int x = e - y * kW;
    int lo = (y - y0) * kWP + x;
    float a = s_rp[lo];
    float t = s_rp[lo + 1];
    float l = s_rp[lo + kWP];
    float all = a + t + l;
    float pv = pos[q];
    run += pv;  // inclusive raster cumsum
    float id_mask = run * pv;
    float Kf = (a > 0.f) ? 1.f : 0.f;
    float Pf = ((t > 0.f) && (l > 0.f)) ? 1.f : 0.f;
    float notK = 1.f - Kf;
    float nv = -(t + l) * Kf - l * Pf * notK + (all + id_mask);
    nv *= dp[e];
    int od = base + (y + 1) * kWP + (x + 1);
    int lod = (y + 1 - y0) * kWP + (x + 1);
    nv = fmaf(nv, s_m[lod], E[od]);
    resN[od] = nv;
    if (nv != res[od]) diff = true;  // last == incoming res

    float PK = Pf * notK;
    if (y >= 1 && x >= 1) {
      int o2 = base + (y + 1) * kWP + x;
      MN[o2] = 1.f - PK;
      EN[o2] = l * PK;
    }
    if (x == 0) {  // column j=0, rows 1..640
      int ob = base + (y + 1) * kWP;
      resN[ob] = 0.f; MN[ob] = 1.f; EN[ob] = 0.f;
    }
    if (x == kW - 1 && y >= 1) {  // column j=640, rows 2..640
      int ob = base + (y + 1) * kWP + kW;
      MN[ob] = 1.f; EN[ob] = 0.f;
    }
    if (y == 0) {  // rows i=0 and i=1
      int ob0 = base + (x + 1);
      int ob1 = base + kWP + (x + 1);
      resN[ob0] = 0.f; MN[ob0] = 1.f; EN[ob0] = 0.f;
      MN[ob1] = 1.f; EN[ob1] = 0.f;
      if (x == 0) { resN[base] = 0.f; MN[base] = 1.f; EN[base] = 0.f; }
    }
  }
  if (diff) anydiff = 1;
  __syncthreads();
  if (threadIdx.x == 0 && anydiff) atomicOr(&flags[1], 1);
}

// ---------------- phase 4: fold "changed" into "done" ----------------
__global__ void k_final(int* __restrict__ flags) {
  if (flags[0] == 0 && flags[1] == 0) flags[0] = 1;
}

// ---------------- output: strip padding ----------------
__global__ void k_output(const float* __restrict__ res, float* __restrict__ out) {
  int idx = blockIdx.x * blockDim.x + threadIdx.x;
  if (idx >= kB * kNPix) return;
  int b = idx / kNPix;
  int r = idx - b * kNPix;
  int y = r / kW;
  int x = r - y * kW;
  out[idx] = res[b * kPlane + (y + 1) * kWP + (x + 1)];
}

extern "C" void kernel_launch(void* const* d_in, const int* in_sizes, int n_in,
                              void* d_out, int out_size, void* d_ws, size_t ws_size,
                              hipStream_t stream) {
  const float* img = (const float*)d_in[0];
  float* w = (float*)d_ws;
  size_t np = (size_t)kB * kPlane;
  float* resA = w;
  float* resB = resA + np;
  float* MA = resB + np;
  float* MB = MA + np;
  float* EA = MB + np;
  float* EB = EA + np;
  float* partials = EB + np;                     // kB * kPartStride
  float* blkoffs = partials + kB * kPartStride;  // kB * kPartStride
  int* flags = (int*)(blkoffs + kB * kPartStride);

  k_init<<<1024, kThreads, 0, stream>>>(img, resA, MA, EA, flags);

  float* res = resA; float* M = MA; float* E = EA;
  float* resN = resB; float* MN = MB; float* EN = EB;
  for (int it = 0; it < kIters; ++it) {
    k_partials<<<kNBlocks, kThreads, 0, stream>>>(res, M, E, img, partials, flags);
    k_scanblk<<<1, 128, 0, stream>>>(partials, blkoffs, flags);
    k_update<<<kNBlocks, kThreads, 0, stream>>>(res, M, E, resN, MN, EN, img,
                                                blkoffs, flags);
    k_final<<<1, 1, 0, stream>>>(flags);
    float* t;
    t = res;  res = resN;  resN = t;
    t = M;    M = MN;      MN = t;
    t = E;    E = EN;      EN = t;
  }
  k_output<<<(kB * kNPix + kThreads - 1) / kThreads, kThreads, 0, stream>>>(
      res, (float*)d_out);
}